// WalshHadamardTransform_14233521619090
// MI455X (gfx1250) — compile-verified
//
#include <hip/hip_runtime.h>
#include <hip/hip_bf16.h>

typedef __attribute__((ext_vector_type(2))) float v2f;
typedef __attribute__((ext_vector_type(8))) float v8f;

#define LDSW 34        // 32x32 tile, padded row stride (dwords): bank-conflict-free + 8B aligned
#define WAVES_PER_BLOCK 8

__device__ __forceinline__ float hval(int i, int j) {
    // Natural-order (Sylvester) Hadamard: H[i][j] = (-1)^popcount(i & j)
    return (__builtin_popcount(i & j) & 1) ? -1.0f : 1.0f;
}

__device__ __forceinline__ v8f wmma4(v2f a, v2f b, v8f c) {
    // D = A(16x4 f32) * B(4x16 f32) + C(16x16 f32)
    return __builtin_amdgcn_wmma_f32_16x16x4_f32(
        false, a, false, b, (short)0, c, false, false);
}

__global__ __launch_bounds__(256)
void wht1024_wmma_kernel(const float* __restrict__ x, float* __restrict__ out, int nrows) {
    __shared__ float lds[WAVES_PER_BLOCK][32 * LDSW];

    const int lane = threadIdx.x & 31;
    const int wid  = threadIdx.x >> 5;
    const int h    = lane >> 4;   // lane half: selects K pair (A/B frags) / M half (C/D)
    const int t    = lane & 15;   // m for A frags, n for B and C/D frags

    // Row-invariant Hadamard fragments.
    // First matmul  T = H32 * X : A-frag hA[i][q], lane(16h+t) holds H[16i+t][4q+2h+v]
    // Second matmul Y = T * Hs  : B-frag hB[j][q], lane(16h+t) holds Hs[4q+2h+v][16j+t]
    //   (H symmetric -> hB = hA * 2^-5, total WHT scale folded here)
    v2f hA[2][8], hB[2][8];
#pragma unroll
    for (int i = 0; i < 2; ++i) {
#pragma unroll
        for (int q = 0; q < 8; ++q) {
            v2f a;
            a.x = hval(16 * i + t, 4 * q + 2 * h + 0);
            a.y = hval(16 * i + t, 4 * q + 2 * h + 1);
            hA[i][q] = a;
            hB[i][q] = a * 0.03125f;   // 2^-5
        }
    }

    float* L = &lds[wid][0];
    const int gw = blockIdx.x * WAVES_PER_BLOCK + wid;
    const int nw = gridDim.x * WAVES_PER_BLOCK;

    for (int row = gw; row < nrows; row += nw) {
        const float* __restrict__ xr = x + (size_t)row * 1024;

        // ---- First matmul: T = H32 * X, X[a][b] = xr[32a + b] ----
        v8f acc[2][2] = {};
#pragma unroll
        for (int q = 0; q < 8; ++q) {
            // B-frag (j,q): lane(16h+t) needs X[4q+2h+v][16j+t] = xr[128q + 64h + 32v + 16j + t]
            const float* p = xr + 128 * q + 64 * h + t;
            v2f xb0, xb1;
            xb0.x = p[0];
            xb0.y = p[32];
            xb1.x = p[16];
            xb1.y = p[48];
            acc[0][0] = wmma4(hA[0][q], xb0, acc[0][0]);
            acc[0][1] = wmma4(hA[0][q], xb1, acc[0][1]);
            acc[1][0] = wmma4(hA[1][q], xb0, acc[1][0]);
            acc[1][1] = wmma4(hA[1][q], xb1, acc[1][1]);
        }

        // ---- D-layout -> LDS (natural 32x32, padded) ----
#pragma unroll
        for (int i = 0; i < 2; ++i)
#pragma unroll
            for (int j = 0; j < 2; ++j)
#pragma unroll
                for (int v = 0; v < 8; ++v)
                    L[(16 * i + 8 * h + v) * LDSW + 16 * j + t] = acc[i][j][v];

        __builtin_amdgcn_fence(__ATOMIC_RELEASE, "wavefront");
        __builtin_amdgcn_wave_barrier();
        __builtin_amdgcn_fence(__ATOMIC_ACQUIRE, "wavefront");

        // ---- Second matmul: Y = T * Hs ----
        v8f acc2[2][2] = {};
#pragma unroll
        for (int q = 0; q < 8; ++q) {
            // A-frag (i,q): lane(16h+t) needs T[16i+t][4q+2h+v]
            v2f a0, a1;
            a0.x = L[(t)      * LDSW + 4 * q + 2 * h + 0];
            a0.y = L[(t)      * LDSW + 4 * q + 2 * h + 1];
            a1.x = L[(16 + t) * LDSW + 4 * q + 2 * h + 0];
            a1.y = L[(16 + t) * LDSW + 4 * q + 2 * h + 1];
            acc2[0][0] = wmma4(a0, hB[0][q], acc2[0][0]);
            acc2[0][1] = wmma4(a0, hB[1][q], acc2[0][1]);
            acc2[1][0] = wmma4(a1, hB[0][q], acc2[1][0]);
            acc2[1][1] = wmma4(a1, hB[1][q], acc2[1][1]);
        }

        __builtin_amdgcn_fence(__ATOMIC_RELEASE, "wavefront");
        __builtin_amdgcn_wave_barrier();
        __builtin_amdgcn_fence(__ATOMIC_ACQUIRE, "wavefront");

        // ---- Store Y (D layout): vgpr v, lane(16h+t), tile(i,j) -> out[row][32*(16i+8h+v) + 16j + t]
        float* __restrict__ orow = out + (size_t)row * 1024;
#pragma unroll
        for (int i = 0; i < 2; ++i)
#pragma unroll
            for (int j = 0; j < 2; ++j)
#pragma unroll
                for (int v = 0; v < 8; ++v)
                    orow[(16 * i + 8 * h + v) * 32 + 16 * j + t] = acc2[i][j][v];
    }
}

extern "C" void kernel_launch(void* const* d_in, const int* in_sizes, int n_in,
                              void* d_out, int out_size, void* d_ws, size_t ws_size,
                              hipStream_t stream) {
    const float* x = (const float*)d_in[0];
    // d_in[1] (W) is unused: the Hadamard matrix is generated in-register (exact ±2^-5).
    float* out = (float*)d_out;
    const int nrows = in_sizes[0] / 1024;   // 32768

    const int blocks = 1024;                // 8192 waves, 4 rows each (grid-stride safe)
    hipLaunchKernelGGL(wht1024_wmma_kernel, dim3(blocks), dim3(256), 0, stream,
                       x, out, nrows);
}